// VectorDiscoConvS2_32160715112851
// MI455X (gfx1250) — compile-verified
//
#include <hip/hip_runtime.h>
#include <hip/hip_bf16.h>
#include <math.h>

// ---------------------------------------------------------------------------
// VectorDiscoConvS2 on gfx1250 (MI455X), all heavy stages as f32 WMMA GEMMs.
// Pipeline: [tables] -> [forward DFT GEMM, 2 f-tiles/wave] ->
//           [band-einsum + channel GEMM (fused, z in LDS)] x3 ->
//           [inverse DFT GEMM + bias, 4 y-tiles/wave].
// Channel contraction is done in the frequency domain (commutes with irfft),
// so only 46K rows are inverse-transformed instead of 645K.
// ---------------------------------------------------------------------------

#define BATCH   2
#define CSI     64
#define CVI     32
#define CSO     64
#define CVO     32
#define NLAT    180
#define NLON    360
#define NF      181     // rfft freqs
#define FP      192     // freq dim padded to 12 WMMA tiles
#define BW      9
#define NCH     128     // packed spectral channels: [0,64)=scalar, [64,96)=u, [96,128)=v
#define ROWS    (BATCH*NCH*NLAT)        // 46080 spectral rows
#define NFT     (FP/16)                 // 12 freq tiles
#define IKP     388                     // padded LDS stride over (i,k)=384
#define PI_F    3.14159265358979323846f

typedef float v2f __attribute__((ext_vector_type(2)));
typedef float v8f __attribute__((ext_vector_type(8)));

// D = A(16x4) * B(4x16) + C, f32 everywhere.
__device__ __forceinline__ v8f wmma4(v2f a, v2f b, v8f c) {
    return __builtin_amdgcn_wmma_f32_16x16x4_f32(
        /*neg_a=*/false, a, /*neg_b=*/false, b,
        /*c_mod=*/(short)0, c, /*reuse_a=*/false, /*reuse_b=*/false);
}

__device__ __forceinline__ void rowdec(int row, int& b, int& c, int& t) {
    b = row / (NCH * NLAT);
    int rem = row - b * (NCH * NLAT);
    c = rem / NLAT;
    t = rem - c * NLAT;
}

// ---------------------------------------------------------------------------
// Kernel 0: DFT / iDFT tables (argument-reduced for f32 accuracy).
// ctab/stab: [FP][NLON]  B-matrix for forward DFT (K=n contiguous per f row)
// iac/ias  : [NLON][FP]  B-matrix for inverse DFT (K=f contiguous per y row)
// Padded f rows/cols are zero so padded spectra stay zero.
// ---------------------------------------------------------------------------
__global__ void k_tables(float* __restrict__ ctab, float* __restrict__ stab,
                         float* __restrict__ iac,  float* __restrict__ ias) {
    int idx = blockIdx.x * blockDim.x + threadIdx.x;
    if (idx >= FP * NLON) return;
    {   // forward tables
        int f = idx / NLON, n = idx - f * NLON;
        float c = 0.f, s = 0.f;
        if (f <= 180) {
            int m = (n * f) % NLON;                 // exact reduction
            float th = (2.0f * PI_F / NLON) * (float)m;
            c =  __builtin_cosf(th);
            s = -__builtin_sinf(th);                // rfft: X = sum x e^{-i...}
        }
        ctab[idx] = c; stab[idx] = s;
    }
    {   // inverse tables
        int y = idx / FP, f = idx - y * FP;
        float c = 0.f, s = 0.f;
        if (f <= 180) {
            float wf = (f == 0 || f == 180) ? (1.0f / NLON) : (2.0f / NLON);
            int m = (f * y) % NLON;
            float th = (2.0f * PI_F / NLON) * (float)m;
            c =  wf * __builtin_cosf(th);
            s = -wf * __builtin_sinf(th);           // x = sum wf(Re cos - Im sin)
        }
        iac[idx] = c; ias[idx] = s;
    }
}

// ---------------------------------------------------------------------------
// Kernel 1: forward DFT as GEMM.  One wave = 16 rows x 2 freq-tiles, four
// accumulators (re/im x 2 tiles) sharing one A load: 1 A + 4 B b64 loads
// feed 4 WMMAs per K-step (K=360).
// ---------------------------------------------------------------------------
__global__ __launch_bounds__(256) void k_fdft(
        const float* __restrict__ xs, const float* __restrict__ xv,
        const float* __restrict__ ctab, const float* __restrict__ stab,
        float* __restrict__ xfre, float* __restrict__ xfim) {
    int wid  = threadIdx.x >> 5;
    int lane = threadIdx.x & 31;
    int l16  = lane & 15;
    int kh   = lane >> 4;                      // K half: lanes 16-31 hold K=2,3
    int koff = 2 * kh;
    int task = blockIdx.x * 8 + wid;           // 2880 row-tiles x 6 f-groups
    int rt = task / (NFT / 2), fg = task - rt * (NFT / 2);

    int row = rt * 16 + l16;
    int b, c, t; rowdec(row, b, c, t);
    const float* xrow;
    if (c < CSI) {
        xrow = xs + ((size_t)(b * CSI + c) * NLAT + t) * NLON;
    } else {
        int c2 = c - CSI, comp = c2 >> 5, ch = c2 & 31;
        xrow = xv + ((size_t)((b * 2 + comp) * CVI + ch) * NLAT + t) * NLON;
    }
    int f0 = fg * 32 + l16;                    // lane's freq column, tile 0
    const float* cb0 = ctab + (size_t)f0 * NLON;
    const float* sb0 = stab + (size_t)f0 * NLON;
    const float* cb1 = cb0 + (size_t)16 * NLON;
    const float* sb1 = sb0 + (size_t)16 * NLON;

    v8f aR0 = {0,0,0,0,0,0,0,0}, aI0 = {0,0,0,0,0,0,0,0};
    v8f aR1 = {0,0,0,0,0,0,0,0}, aI1 = {0,0,0,0,0,0,0,0};
    for (int n0 = 0; n0 < NLON; n0 += 4) {
        v2f a   = *(const v2f*)(xrow + n0 + koff);
        v2f bc0 = *(const v2f*)(cb0  + n0 + koff);
        v2f bs0 = *(const v2f*)(sb0  + n0 + koff);
        v2f bc1 = *(const v2f*)(cb1  + n0 + koff);
        v2f bs1 = *(const v2f*)(sb1  + n0 + koff);
        aR0 = wmma4(a, bc0, aR0);
        aI0 = wmma4(a, bs0, aI0);
        aR1 = wmma4(a, bc1, aR1);
        aI1 = wmma4(a, bs1, aI1);
    }
    #pragma unroll
    for (int r = 0; r < 8; ++r) {
        int m = r + kh * 8;
        size_t off = (size_t)(rt * 16 + m) * FP + fg * 32 + l16;
        xfre[off]      = aR0[r];
        xfim[off]      = aI0[r];
        xfre[off + 16] = aR1[r];
        xfim[off + 16] = aI1[r];
    }
}

// ---------------------------------------------------------------------------
// Kernel 2: out_s spectra.  Per block (b,t,f-tile):
//   build z[(i,k)=384][32 cols(re0..15,im0..15)] in LDS (band einsum, VALU),
//   then GEMM  OF[o] = [w_ss | w_vs] @ z  with 8 waves x 96 WMMA.
// z rows: [0,256)=ss band (s_gather, psi_s); [256,384)=vs band
//         (v_gather, U*psi_vcos + V*psi_vsin).
// ---------------------------------------------------------------------------
__global__ __launch_bounds__(256) void k_outs(
        const float* __restrict__ xfre, const float* __restrict__ xfim,
        const float* __restrict__ psi_s, const int* __restrict__ sg,
        const float* __restrict__ psi_vc, const float* __restrict__ psi_vs,
        const int* __restrict__ vg,
        const float* __restrict__ wss, const float* __restrict__ wvs,
        float* __restrict__ ofre, float* __restrict__ ofim) {
    __shared__ float z[32 * IKP];

    int blk = blockIdx.x;
    int ft = blk % NFT;
    int t  = (blk / NFT) % NLAT;
    int b  = blk / (NFT * NLAT);

    for (int task = threadIdx.x; task < 384 * 16; task += 256) {
        int ik = task >> 4;
        int fl = task & 15;
        int f  = ft * 16 + fl;
        float zr = 0.f, zi = 0.f;
        if (f <= 180) {
            if (ik < 256) {                       // scalar band, psi_s
                int i = ik >> 2, k = ik & 3;
                #pragma unroll
                for (int w = 0; w < BW; ++w) {
                    int g = sg[t * BW + w];
                    size_t xo = ((size_t)(b * NCH + i) * NLAT + g) * FP + f;
                    float xr = xfre[xo], xi = xfim[xo];
                    size_t po = ((((size_t)k * NLAT + t) * BW + w) * NF + f) * 2;
                    float pr = psi_s[po], pi = psi_s[po + 1];
                    zr += xr * pr - xi * pi;
                    zi += xr * pi + xi * pr;
                }
            } else {                              // vs band: U*cos + V*sin
                int ik2 = ik - 256;
                int i = ik2 >> 2, k = ik2 & 3;
                #pragma unroll
                for (int w = 0; w < BW; ++w) {
                    int g = vg[t * BW + w];
                    size_t uo = ((size_t)(b * NCH + 64 + i) * NLAT + g) * FP + f;
                    size_t vo = ((size_t)(b * NCH + 96 + i) * NLAT + g) * FP + f;
                    float ur = xfre[uo], ui = xfim[uo];
                    float vr = xfre[vo], vi = xfim[vo];
                    size_t po = ((((size_t)k * NLAT + t) * BW + w) * NF + f) * 2;
                    float cr = psi_vc[po], ci = psi_vc[po + 1];
                    float sr = psi_vs[po], si = psi_vs[po + 1];
                    zr += ur * cr - ui * ci + vr * sr - vi * si;
                    zi += ur * ci + ui * cr + vr * si + vi * sr;
                }
            }
        }
        z[fl * IKP + ik]        = zr;
        z[(fl + 16) * IKP + ik] = zi;
    }
    __syncthreads();

    // GEMM: 8 waves -> (Mtile 0..3) x (re/im)
    int wid  = threadIdx.x >> 5;
    int lane = threadIdx.x & 31;
    int l16  = lane & 15, kh = lane >> 4, koff = 2 * kh;
    int o0 = (wid >> 1) * 16;
    int n0 = (wid & 1) * 16;
    int orow = o0 + l16;
    v8f acc = {0,0,0,0,0,0,0,0};
    const float* arow0 = wss + (size_t)orow * 256;
    const float* arow1 = wvs + (size_t)orow * 128;
    for (int kk = 0; kk < 256; kk += 4) {
        v2f a = *(const v2f*)(arow0 + kk + koff);
        v2f bb = *(const v2f*)(&z[(n0 + l16) * IKP + kk + koff]);
        acc = wmma4(a, bb, acc);
    }
    for (int kk = 0; kk < 128; kk += 4) {
        v2f a = *(const v2f*)(arow1 + kk + koff);
        v2f bb = *(const v2f*)(&z[(n0 + l16) * IKP + 256 + kk + koff]);
        acc = wmma4(a, bb, acc);
    }
    float* of = (wid & 1) ? ofim : ofre;
    #pragma unroll
    for (int r = 0; r < 8; ++r) {
        int c = o0 + r + kh * 8;                  // output channel 0..63
        size_t off = ((size_t)(b * NCH + c) * NLAT + t) * FP + ft * 16 + l16;
        of[off] = acc[r];
    }
}

// ---------------------------------------------------------------------------
// Kernel 3: out_v spectra, one plane per launch (0=theta, 1=phi).
// z rows: [0,256) = sv band (XS, v_gather, psi_vcos|psi_vsin)
//         [256,384) = vv band (s_gather):
//            theta: U*psi_scos - V*psi_ssin ; phi: U*psi_ssin + V*psi_scos
// GEMM A = [w_sv | w_vv], output channels 64+plane*32 .. +31.
// ---------------------------------------------------------------------------
__global__ __launch_bounds__(128) void k_outv(
        const float* __restrict__ xfre, const float* __restrict__ xfim,
        const float* __restrict__ psi_vc, const float* __restrict__ psi_vs,
        const int* __restrict__ vg,
        const float* __restrict__ psi_sc, const float* __restrict__ psi_ss,
        const int* __restrict__ sg,
        const float* __restrict__ wsv, const float* __restrict__ wvv,
        float* __restrict__ ofre, float* __restrict__ ofim, int plane) {
    __shared__ float z[32 * IKP];

    int blk = blockIdx.x;
    int ft = blk % NFT;
    int t  = (blk / NFT) % NLAT;
    int b  = blk / (NFT * NLAT);

    for (int task = threadIdx.x; task < 384 * 16; task += 128) {
        int ik = task >> 4;
        int fl = task & 15;
        int f  = ft * 16 + fl;
        float zr = 0.f, zi = 0.f;
        if (f <= 180) {
            if (ik < 256) {                       // sv band: XS with psi_v
                int i = ik >> 2, k = ik & 3;
                const float* psi = plane ? psi_vs : psi_vc;
                #pragma unroll
                for (int w = 0; w < BW; ++w) {
                    int g = vg[t * BW + w];
                    size_t xo = ((size_t)(b * NCH + i) * NLAT + g) * FP + f;
                    float xr = xfre[xo], xi = xfim[xo];
                    size_t po = ((((size_t)k * NLAT + t) * BW + w) * NF + f) * 2;
                    float pr = psi[po], pi = psi[po + 1];
                    zr += xr * pr - xi * pi;
                    zi += xr * pi + xi * pr;
                }
            } else {                              // vv band: U,V with psi_s cos/sin
                int ik2 = ik - 256;
                int i = ik2 >> 2, k = ik2 & 3;
                #pragma unroll
                for (int w = 0; w < BW; ++w) {
                    int g = sg[t * BW + w];
                    size_t uo = ((size_t)(b * NCH + 64 + i) * NLAT + g) * FP + f;
                    size_t vo = ((size_t)(b * NCH + 96 + i) * NLAT + g) * FP + f;
                    float ur = xfre[uo], ui = xfim[uo];
                    float vr = xfre[vo], vi = xfim[vo];
                    size_t po = ((((size_t)k * NLAT + t) * BW + w) * NF + f) * 2;
                    float cr = psi_sc[po], ci = psi_sc[po + 1];
                    float sr = psi_ss[po], si = psi_ss[po + 1];
                    if (plane == 0) {             // U*c - V*s
                        zr += ur * cr - ui * ci - (vr * sr - vi * si);
                        zi += ur * ci + ui * cr - (vr * si + vi * sr);
                    } else {                      // U*s + V*c
                        zr += ur * sr - ui * si + vr * cr - vi * ci;
                        zi += ur * si + ui * sr + vr * ci + vi * cr;
                    }
                }
            }
        }
        z[fl * IKP + ik]        = zr;
        z[(fl + 16) * IKP + ik] = zi;
    }
    __syncthreads();

    // GEMM: 4 waves -> (Mtile 0..1) x (re/im)
    int wid  = threadIdx.x >> 5;
    int lane = threadIdx.x & 31;
    int l16  = lane & 15, kh = lane >> 4, koff = 2 * kh;
    int o0 = (wid >> 1) * 16;
    int n0 = (wid & 1) * 16;
    int orow = o0 + l16;
    v8f acc = {0,0,0,0,0,0,0,0};
    const float* arow0 = wsv + (size_t)orow * 256;
    const float* arow1 = wvv + (size_t)orow * 128;
    for (int kk = 0; kk < 256; kk += 4) {
        v2f a = *(const v2f*)(arow0 + kk + koff);
        v2f bb = *(const v2f*)(&z[(n0 + l16) * IKP + kk + koff]);
        acc = wmma4(a, bb, acc);
    }
    for (int kk = 0; kk < 128; kk += 4) {
        v2f a = *(const v2f*)(arow1 + kk + koff);
        v2f bb = *(const v2f*)(&z[(n0 + l16) * IKP + 256 + kk + koff]);
        acc = wmma4(a, bb, acc);
    }
    float* of = (wid & 1) ? ofim : ofre;
    #pragma unroll
    for (int r = 0; r < 8; ++r) {
        int c = 64 + plane * 32 + o0 + r + kh * 8;
        size_t off = ((size_t)(b * NCH + c) * NLAT + t) * FP + ft * 16 + l16;
        of[off] = acc[r];
    }
}

// ---------------------------------------------------------------------------
// Kernel 4: inverse DFT as GEMM + bias.  One wave = 16 rows x 4 lon-tiles
// (64 lon points): 2 A loads feed 8 WMMAs per K-step (K=192; padded spectra
// are zero).  iDFT tables (0.5 MB) stay L2-resident.  Writes d_out.
// ---------------------------------------------------------------------------
__global__ __launch_bounds__(256) void k_idft(
        const float* __restrict__ ofre, const float* __restrict__ ofim,
        const float* __restrict__ iac, const float* __restrict__ ias,
        const float* __restrict__ bias_s, const float* __restrict__ bias_v,
        float* __restrict__ out) {
    const int NYG = 6;                            // 6 groups of 4 y-tiles (384 cols)
    int wid  = threadIdx.x >> 5;
    int lane = threadIdx.x & 31;
    int l16  = lane & 15, kh = lane >> 4, koff = 2 * kh;
    int task = blockIdx.x * 8 + wid;              // 2880 x 6 = 17280 waves
    int rt = task / NYG, yg = task - rt * NYG;

    int row = rt * 16 + l16;
    const float* are = ofre + (size_t)row * FP;
    const float* aim = ofim + (size_t)row * FP;
    int ybase = yg * 64;

    const float* bcp[4];
    const float* bsp[4];
    #pragma unroll
    for (int j = 0; j < 4; ++j) {
        int y  = ybase + j * 16 + l16;
        int yc = (y < NLON) ? y : (NLON - 1);     // clamp B loads; store guarded
        bcp[j] = iac + (size_t)yc * FP;
        bsp[j] = ias + (size_t)yc * FP;
    }

    v8f acc[4];
    #pragma unroll
    for (int j = 0; j < 4; ++j) acc[j] = (v8f){0,0,0,0,0,0,0,0};

    for (int k0 = 0; k0 < FP; k0 += 4) {
        v2f ar = *(const v2f*)(are + k0 + koff);
        v2f ai = *(const v2f*)(aim + k0 + koff);
        #pragma unroll
        for (int j = 0; j < 4; ++j) {
            v2f tc = *(const v2f*)(bcp[j] + k0 + koff);
            v2f ts = *(const v2f*)(bsp[j] + k0 + koff);
            acc[j] = wmma4(ar, tc, acc[j]);       // Re * wf cos
            acc[j] = wmma4(ai, ts, acc[j]);       // Im * (-wf sin)
        }
    }

    #pragma unroll
    for (int r = 0; r < 8; ++r) {
        int m = r + kh * 8;
        int rw = rt * 16 + m;
        int b, c, t; rowdec(rw, b, c, t);
        float bv;
        size_t off;
        if (c < CSO) {
            bv = bias_s[c];
            off = ((size_t)(b * CSO + c) * NLAT + t) * NLON;
        } else {
            int c2 = c - 64, comp = c2 >> 5, o = c2 & 31;
            bv = bias_v[o];
            off = (size_t)BATCH * CSO * NLAT * NLON +
                  ((size_t)((b * 2 + comp) * CVO + o) * NLAT + t) * NLON;
        }
        #pragma unroll
        for (int j = 0; j < 4; ++j) {
            int y = ybase + j * 16 + l16;
            if (y < NLON) out[off + y] = acc[j][r] + bv;
        }
    }
}

// ---------------------------------------------------------------------------
// Host launcher: d_in order follows setup_inputs(); complex64 arrays are
// interleaved (re,im) float pairs.  Workspace layout (floats):
//   ctab 69120 | stab 69120 | iac 69120 | ias 69120 |
//   xfre 8847360 | xfim 8847360 | ofre 8847360 | ofim 8847360   (~143 MB)
// ---------------------------------------------------------------------------
extern "C" void kernel_launch(void* const* d_in, const int* in_sizes, int n_in,
                              void* d_out, int out_size, void* d_ws, size_t ws_size,
                              hipStream_t stream) {
    const float* xs     = (const float*)d_in[0];
    const float* xv     = (const float*)d_in[1];
    const float* psi_s  = (const float*)d_in[2];
    const float* psi_sc = (const float*)d_in[3];
    const float* psi_ss = (const float*)d_in[4];
    const int*   sg     = (const int*)  d_in[5];
    const float* psi_vc = (const float*)d_in[6];
    const float* psi_vs = (const float*)d_in[7];
    const int*   vg     = (const int*)  d_in[8];
    const float* wss    = (const float*)d_in[9];
    const float* wvs    = (const float*)d_in[10];
    const float* wsv    = (const float*)d_in[11];
    const float* wvv    = (const float*)d_in[12];
    const float* bs     = (const float*)d_in[13];
    const float* bv     = (const float*)d_in[14];
    float* out = (float*)d_out;

    float* ws   = (float*)d_ws;
    float* ctab = ws;
    float* stab = ctab + (size_t)FP * NLON;
    float* iac  = stab + (size_t)FP * NLON;
    float* ias  = iac  + (size_t)NLON * FP;
    float* xfre = ias  + (size_t)NLON * FP;
    float* xfim = xfre + (size_t)ROWS * FP;
    float* ofre = xfim + (size_t)ROWS * FP;
    float* ofim = ofre + (size_t)ROWS * FP;

    const int NBLK_SPEC = BATCH * NLAT * NFT;     // 4320

    k_tables<<<(FP * NLON + 255) / 256, 256, 0, stream>>>(ctab, stab, iac, ias);
    k_fdft  <<<(ROWS / 16) * (NFT / 2) / 8, 256, 0, stream>>>(xs, xv, ctab, stab,
                                                              xfre, xfim);
    k_outs  <<<NBLK_SPEC, 256, 0, stream>>>(xfre, xfim, psi_s, sg, psi_vc, psi_vs, vg,
                                            wss, wvs, ofre, ofim);
    k_outv  <<<NBLK_SPEC, 128, 0, stream>>>(xfre, xfim, psi_vc, psi_vs, vg,
                                            psi_sc, psi_ss, sg, wsv, wvv, ofre, ofim, 0);
    k_outv  <<<NBLK_SPEC, 128, 0, stream>>>(xfre, xfim, psi_vc, psi_vs, vg,
                                            psi_sc, psi_ss, sg, wsv, wvv, ofre, ofim, 1);
    k_idft  <<<(ROWS / 16) * 6 / 8, 256, 0, stream>>>(ofre, ofim, iac, ias, bs, bv, out);
}